// Gate_3891240370244
// MI455X (gfx1250) — compile-verified
//
#include <hip/hip_runtime.h>
#include <hip/hip_bf16.h>

typedef __attribute__((ext_vector_type(2))) float v2f;
typedef __attribute__((ext_vector_type(8))) float v8f;
typedef __attribute__((ext_vector_type(4))) unsigned int u32x4;
typedef __attribute__((ext_vector_type(8))) int i32x8;
typedef __attribute__((ext_vector_type(4))) int i32x4;

#define T_TOKENS 32768
#define DIM      2048
#define KDIM     2047   // features used: x[:, 1:]
#define NEXP     64
#define NGRP     8
#define TOPK     8
#define TOPKG    4
#define ROUTE_SCALE 2.5f

#define KC        64            // K-chunk staged in LDS
#define NCHUNK    32            // 2047 -> 32 chunks of 64 (last padded via OOB zeros)
#define XSTR      68            // x tile LDS stride (floats) -> conflict-free b64 reads
#define WSTR      68            // weight tile LDS stride
#define SSTR      65            // score LDS stride
#define TOK_BLK   128           // tokens per block (8 waves x 16)
#define THREADS   256

#define BUFX (TOK_BLK * XSTR)   // 8704 floats
#define BUFW (NEXP   * WSTR)    // 4352 floats
#define BUFSZ (BUFX + BUFW)     // 13056 floats per buffer

// Issue one 2D TDM tile load: rows x 64 elements (f32), row pitch
// `strideElems`, LDS destination padded 64+4 DWORDs per row (stride 68).
// Columns >= validCols read as zero (handles the K=2047 tail column).
__device__ __forceinline__ void tdm_load_tile(const float* gaddr, unsigned ldsByte,
                                              unsigned rows, unsigned validCols,
                                              unsigned long long strideElems)
{
    unsigned long long ga = (unsigned long long)(uintptr_t)gaddr;
    u32x4 g0;
    g0[0] = 1u;                                              // count=1 (valid, user mode)
    g0[1] = ldsByte;                                         // lds_addr
    g0[2] = (unsigned)(ga & 0xffffffffu);                    // global_addr[31:0]
    g0[3] = (unsigned)((ga >> 32) & 0x1ffffffu) | (2u << 30);// global_addr[56:32] | type=2
    i32x8 g1;
    g1[0] = (2 << 16)        // data_size = 4 bytes
          | (1 << 20)        // pad_enable
          | (5 << 22)        // pad_interval: every 64 DWORDs
          | (3 << 25);       // pad_amount: 4 DWORDs  -> LDS row stride 68
    g1[1] = (int)((validCols & 0xffffu) << 16);              // tensor_dim0[15:0]
    g1[2] = (int)(((validCols >> 16) & 0xffffu) | ((rows & 0xffffu) << 16)); // dim0 hi | dim1 lo
    g1[3] = (int)(KC << 16);                                 // dim1 hi=0 | tile_dim0=64
    g1[4] = (int)(rows & 0xffffu);                           // tile_dim1=rows | tile_dim2=0
    g1[5] = (int)(strideElems & 0xffffffffu);                // tensor_dim0_stride lo32
    g1[6] = (int)((strideElems >> 32) & 0xffffu);            // dim0_stride hi16 | dim1_stride=0
    g1[7] = 0;
    i32x4 z4 = {0, 0, 0, 0};
    i32x8 z8 = {0, 0, 0, 0, 0, 0, 0, 0};
    // clang-23 / therock-10.0 6-arg form (extra int32x8 unused for 2D tiles)
    __builtin_amdgcn_tensor_load_to_lds(g0, g1, z4, z4, z8, 0);
}

__global__ __launch_bounds__(THREADS)
void moe_gate_kernel(const float* __restrict__ x,
                     const float* __restrict__ w,
                     const float* __restrict__ bias,
                     float* __restrict__ out_w,
                     int*   __restrict__ out_i)
{
    // Double-buffered tiles (2 x 52.2KB). Scores/bias alias buffer 0 afterwards.
    __shared__ __align__(16) float smem[2 * BUFSZ];
    float* smS = smem;           // scores [TOK_BLK][SSTR] (fits in buf0 x region)
    float* smB = smem + BUFX;    // bias   [NEXP]          (buf0 w region)

    const int tid  = threadIdx.x;
    const int lane = tid & 31;
    const int wave = tid >> 5;
    const int blockTok = blockIdx.x * TOK_BLK;

    const int m    = lane & 15;        // M (token) / N (expert) index within fragment
    const int ksel = (lane >> 4) << 1; // lanes 16-31 carry K+2, K+3
    const int waveTok = wave * 16;

    const float* xblk = x + (size_t)blockTok * DIM + 1;   // feature offset +1

    v8f acc0 = {}, acc1 = {}, acc2 = {}, acc3 = {};

    // Prologue: DMA chunk 0 into buffer 0 (wave 0 issues; EXEC ignored by TDM)
    if (wave == 0) {
        tdm_load_tile(xblk, (unsigned)(uintptr_t)&smem[0],
                      TOK_BLK, (unsigned)KDIM, DIM);
        tdm_load_tile(w, (unsigned)(uintptr_t)&smem[BUFX],
                      NEXP, (unsigned)KDIM, KDIM);
    }

    for (int c = 0; c < NCHUNK; ++c) {
        const int cur = c & 1;
        __syncthreads();   // everyone done reading buf[1-cur] (chunk c-1's tiles)
        if (c + 1 < NCHUNK) {
            if (wave == 0) {
                int kc = (c + 1) * KC;
                float* nb = smem + (1 - cur) * BUFSZ;
                tdm_load_tile(xblk + kc, (unsigned)(uintptr_t)nb,
                              TOK_BLK, (unsigned)(KDIM - kc), DIM);
                tdm_load_tile(w + kc, (unsigned)(uintptr_t)(nb + BUFX),
                              NEXP, (unsigned)(KDIM - kc), KDIM);
            }
            __builtin_amdgcn_s_wait_tensorcnt((short)2);  // chunk c's 2 ops retired
        } else {
            __builtin_amdgcn_s_wait_tensorcnt((short)0);
        }
        __syncthreads();   // chunk c's tiles visible to all waves

        const float* base = smem + cur * BUFSZ;
        const float* ax = base + (waveTok + m) * XSTR + ksel;
        const float* bw = base + BUFX + m * WSTR + ksel;
        #pragma unroll
        for (int kk = 0; kk < KC; kk += 4) {
            v2f a  = *(const v2f*)(ax + kk);
            v2f b0 = *(const v2f*)(bw + kk);
            v2f b1 = *(const v2f*)(bw + 16 * WSTR + kk);
            v2f b2 = *(const v2f*)(bw + 32 * WSTR + kk);
            v2f b3 = *(const v2f*)(bw + 48 * WSTR + kk);
            acc0 = __builtin_amdgcn_wmma_f32_16x16x4_f32(false, a, false, b0, (short)0, acc0, false, false);
            acc1 = __builtin_amdgcn_wmma_f32_16x16x4_f32(false, a, false, b1, (short)0, acc1, false, false);
            acc2 = __builtin_amdgcn_wmma_f32_16x16x4_f32(false, a, false, b2, (short)0, acc2, false, false);
            acc3 = __builtin_amdgcn_wmma_f32_16x16x4_f32(false, a, false, b3, (short)0, acc3, false, false);
        }
    }
    // Last chunk read buf1; buf0 is free (last touched at chunk 30, and the
    // c=31 top-of-loop barrier ordered that) -> write scores/bias into buf0.

    // --- epilogue: sigmoid(logits) -> LDS score tile ---
    const int mrow = (lane >> 4) << 3;             // C rows: lanes 16-31 hold M+8
    #pragma unroll
    for (int r = 0; r < 8; ++r) {
        int tok = waveTok + mrow + r;
        float v0 = acc0[r], v1 = acc1[r], v2 = acc2[r], v3 = acc3[r];
        smS[tok * SSTR +  0 + m] = 1.0f / (1.0f + __expf(-v0));
        smS[tok * SSTR + 16 + m] = 1.0f / (1.0f + __expf(-v1));
        smS[tok * SSTR + 32 + m] = 1.0f / (1.0f + __expf(-v2));
        smS[tok * SSTR + 48 + m] = 1.0f / (1.0f + __expf(-v3));
    }
    if (tid < NEXP) smB[tid] = bias[tid];
    __syncthreads();

    // --- routing: one thread per token ---
    if (tid < TOK_BLK) {
        const float* s = &smS[tid * SSTR];

        // group score = sum of top-2 (scores + bias) per group of 8
        float gs[NGRP];
        #pragma unroll
        for (int g = 0; g < NGRP; ++g) {
            float m1 = -INFINITY, m2 = -INFINITY;
            #pragma unroll
            for (int j = 0; j < 8; ++j) {
                float v = s[g * 8 + j] + smB[g * 8 + j];
                if (v > m1)      { m2 = m1; m1 = v; }
                else if (v > m2) { m2 = v; }
            }
            gs[g] = m1 + m2;
        }
        // top-4 groups (strict > == lax.top_k stable lowest-index tie-break)
        unsigned keepg = 0;
        for (int i = 0; i < TOPKG; ++i) {
            int bg = 0; float bv = -INFINITY;
            for (int g = 0; g < NGRP; ++g)
                if (!((keepg >> g) & 1u) && gs[g] > bv) { bv = gs[g]; bg = g; }
            keepg |= 1u << bg;
        }
        // top-8 experts over kept groups
        unsigned long long picked = 0;
        float wsel[TOPK]; int isel[TOPK]; float wsum = 0.0f;
        for (int i = 0; i < TOPK; ++i) {
            int be = 0; float bv = -INFINITY;
            for (int e = 0; e < NEXP; ++e) {
                if (!((keepg >> (e >> 3)) & 1u)) continue;
                if ((picked >> e) & 1ull) continue;
                float v = s[e] + smB[e];
                if (v > bv) { bv = v; be = e; }
            }
            picked |= 1ull << be;
            float sc = s[be];                 // original sigmoid score
            wsel[i] = sc; isel[i] = be; wsum += sc;
        }
        float scale = ROUTE_SCALE / wsum;
        long long tg = blockTok + tid;
        #pragma unroll
        for (int i = 0; i < TOPK; ++i) {
            out_w[tg * TOPK + i] = wsel[i] * scale;
            out_i[tg * TOPK + i] = isel[i];
        }
    }
}

extern "C" void kernel_launch(void* const* d_in, const int* in_sizes, int n_in,
                              void* d_out, int out_size, void* d_ws, size_t ws_size,
                              hipStream_t stream) {
    (void)in_sizes; (void)n_in; (void)out_size; (void)d_ws; (void)ws_size;
    const float* x    = (const float*)d_in[0];
    const float* w    = (const float*)d_in[1];
    const float* bias = (const float*)d_in[2];
    float* out_w = (float*)d_out;                                   // (T, 8) f32
    int*   out_i = (int*)((float*)d_out + (size_t)T_TOKENS * TOPK); // (T, 8) i32

    dim3 grid(T_TOKENS / TOK_BLK);
    moe_gate_kernel<<<grid, THREADS, 0, stream>>>(x, w, bias, out_w, out_i);
}